// Regression_14370960573225
// MI455X (gfx1250) — compile-verified
//
#include <hip/hip_runtime.h>
#include <hip/hip_bf16.h>

// Problem geometry (fixed by the reference).
#define D_MAX   192
#define HGT     384
#define WID     768
#define HW      (HGT * WID)          // 294912 floats between consecutive d-slices
#define CH      16                   // d-rows staged per chunk
#define NCHUNK  (D_MAX / CH)         // 12
#define WAVES   8
#define BLOCK   (WAVES * 32)         // 256 threads, wave32

// Convert a generic pointer to a __shared__ object into the raw 32-bit LDS
// byte offset that VDS / async-LDS instructions expect.
typedef __attribute__((address_space(3))) const void* lds_cptr_t;
__device__ __forceinline__ unsigned lds_off_u32(const void* p) {
    return (unsigned)(unsigned long long)(lds_cptr_t)p;
}

__global__ __launch_bounds__(BLOCK) void topk3_softmax_disp(
        const float* __restrict__ cost, float* __restrict__ out, int npix4) {
    // [buf][wave][row][lane] : 2 * 8 * 16 * 32 * 16B = 128 KB
    __shared__ float4 stage[2][WAVES][CH][32];

    const int t = blockIdx.x * BLOCK + threadIdx.x;
    if (t >= npix4) return;                     // never taken (sizes divide exactly)
    const int wv = threadIdx.x >> 5;
    const int ln = threadIdx.x & 31;

    const int p4 = t << 2;                      // first of 4 pixels owned by this thread
    const int b  = p4 / HW;                     // waves never straddle a batch (HW % 128 == 0)
    const int hw = p4 - b * HW;
    // byte offset of (b, d=0, hw) from `cost`; max < 2^31 so it is a valid
    // signed 32-bit voffset for the saddr-form async load.
    const unsigned gbyte0 = ((unsigned)b * (unsigned)(D_MAX * HW) + (unsigned)hw) * 4u;

    float v0[4], v1[4], v2[4], i0[4], i1[4], i2[4];
#pragma unroll
    for (int k = 0; k < 4; ++k) {
        v0[k] = v1[k] = v2[k] = -INFINITY;
        i0[k] = i1[k] = i2[k] = 0.0f;
    }

    // Issue CH async B128 copies (one per d-row) into buffer BUF.
    // Each lane moves 16B; a wave moves 512B/row, fully coalesced.
#define ISSUE_CHUNK(BUF, C)                                                    \
    do {                                                                       \
        _Pragma("unroll")                                                      \
        for (int r = 0; r < CH; ++r) {                                         \
            unsigned lo = lds_off_u32(&stage[(BUF)][wv][r][ln]);               \
            unsigned go = gbyte0 + (unsigned)((((C) * CH) + r) * HW) * 4u;     \
            asm volatile("global_load_async_to_lds_b128 %0, %1, %2"            \
                         :: "v"(lo), "v"(go), "s"(cost)                        \
                         : "memory");                                          \
        }                                                                      \
    } while (0)

    ISSUE_CHUNK(0, 0);                          // prologue: chunk 0 in flight

    for (int c = 0; c < NCHUNK; ++c) {
        if (c + 1 < NCHUNK) {
            // Buffer (c+1)&1 was last read in the previous iteration; make sure those
            // ds_loads retired before the async engine overwrites it.
            asm volatile("s_wait_dscnt 0x0" ::: "memory");
            ISSUE_CHUNK((c + 1) & 1, c + 1);    // now 32 async ops outstanding
            // Async loads complete in order: <=16 outstanding => chunk c landed.
            asm volatile("s_wait_asynccnt 0x10" ::: "memory");
        } else {
            asm volatile("s_wait_asynccnt 0x0" ::: "memory");
        }

        const int buf = c & 1;
#pragma unroll
        for (int r = 0; r < CH; ++r) {
            const float4 q = stage[buf][wv][r][ln];   // ds_load_b128
            const float fd = (float)(c * CH + r);     // disparity index (exact)
            const float qv[4] = {q.x, q.y, q.z, q.w};
#pragma unroll
            for (int k = 0; k < 4; ++k) {
                const float v  = qv[k];
                const bool  g0 = v > v0[k];           // strict > keeps lower index
                const bool  g1 = v > v1[k];           // on ties (top_k stability)
                const bool  g2 = v > v2[k];
                v2[k] = g1 ? v1[k] : (g2 ? v  : v2[k]);
                i2[k] = g1 ? i1[k] : (g2 ? fd : i2[k]);
                v1[k] = g0 ? v0[k] : (g1 ? v  : v1[k]);
                i1[k] = g0 ? i0[k] : (g1 ? fd : i1[k]);
                v0[k] = g0 ? v  : v0[k];
                i0[k] = g0 ? fd : i0[k];
            }
        }
    }
#undef ISSUE_CHUNK

    // softmax over the 3 survivors (shifted by the max, so e0 == 1).
    float acc[4];
#pragma unroll
    for (int k = 0; k < 4; ++k) {
        const float e1 = __expf(v1[k] - v0[k]);
        const float e2 = __expf(v2[k] - v0[k]);
        const float s  = 1.0f + e1 + e2;
        acc[k] = (i0[k] + i1[k] * e1 + i2[k] * e2) / s;
    }
    float4 res;
    res.x = acc[0]; res.y = acc[1]; res.z = acc[2]; res.w = acc[3];
    reinterpret_cast<float4*>(out)[t] = res;    // [B,1,H,W] flat == pixel order
}

extern "C" void kernel_launch(void* const* d_in, const int* in_sizes, int n_in,
                              void* d_out, int out_size, void* d_ws, size_t ws_size,
                              hipStream_t stream) {
    const float* cost = (const float*)d_in[0];    // [2,192,384,768] fp32
    float*       out  = (float*)d_out;            // [2,1,384,768]   fp32
    const int npix4  = out_size / 4;              // 147456 threads, 4 pixels each
    const int blocks = (npix4 + BLOCK - 1) / BLOCK; // 576 blocks of 256
    topk3_softmax_disp<<<blocks, BLOCK, 0, stream>>>(cost, out, npix4);
}